// VectorQuantizer_5317169513002
// MI455X (gfx1250) — compile-verified
//
#include <hip/hip_runtime.h>
#include <hip/hip_bf16.h>

// ---- problem constants (match reference) ----
constexpr int   K_CODES        = 512;
constexpr int   DIM            = 64;
constexpr int   N_TOTAL        = 131072;       // 32 * 64 * 64 rows (NHWC-flattened)
constexpr int   HW             = 4096;         // 64*64
constexpr int   ROWS_PER_WAVE  = 16;
constexpr int   WAVES          = 8;            // 256 threads = 8 wave32
constexpr int   ROWS_PER_BLOCK = ROWS_PER_WAVE * WAVES;   // 128
constexpr int   NBLOCKS        = N_TOTAL / ROWS_PER_BLOCK; // 1024

constexpr int   CB_STRIDE = 65;   // 64 data cols + pad (bank-conflict free)
constexpr int   X_STRIDE  = 65;   // 64 data cols + row-norm in col 64

// output layout (flat concatenation in return order: loss, quantized, encodings, indices)
constexpr size_t Q_OFF   = 1;
constexpr size_t ENC_OFF = Q_OFF + (size_t)N_TOTAL * DIM;       // 1 + 8388608
constexpr size_t IDX_OFF = ENC_OFF + (size_t)N_TOTAL * K_CODES; // + 67108864

// shared memory layout (floats)
constexpr int CB_FLOATS  = K_CODES * CB_STRIDE;                   // 33280
constexpr int XT_FLOATS  = WAVES * ROWS_PER_WAVE * X_STRIDE;      // 8320
constexpr int SMEM_FLOATS = CB_FLOATS + K_CODES /*enorm*/ + XT_FLOATS
                          + WAVES * ROWS_PER_WAVE /*idx (int)*/ + WAVES /*wloss*/;
constexpr size_t SMEM_BYTES = (size_t)SMEM_FLOATS * 4;

typedef float v2f __attribute__((ext_vector_type(2)));
typedef float v8f __attribute__((ext_vector_type(8)));

// Load 16 B fragments (one 16-code column tile, K=64) into registers.
#define LOAD_B(ct, dst) do {                                             \
    const float* cbt_ = cb_lds + ((ct) * 16 + l16) * CB_STRIDE;          \
    _Pragma("unroll")                                                    \
    for (int kk = 0; kk < 16; ++kk) {                                    \
        int k0 = 4 * kk + 2 * half;                                      \
        (dst)[kk].x = cbt_[k0];                                          \
        (dst)[kk].y = cbt_[k0 + 1];                                      \
    }                                                                    \
} while (0)

// 16 back-to-back fp32 WMMAs over K=64, then distance + running argmin.
#define DO_TILE(ct, bsrc) do {                                           \
    v8f acc_ = {0.f, 0.f, 0.f, 0.f, 0.f, 0.f, 0.f, 0.f};                 \
    _Pragma("unroll")                                                    \
    for (int kk = 0; kk < 16; ++kk)                                      \
        acc_ = __builtin_amdgcn_wmma_f32_16x16x4_f32(                    \
            false, afrag[kk], false, (bsrc)[kk], (short)0, acc_,         \
            false, false);                                               \
    const int ncol_ = (ct) * 16 + l16;                                   \
    const float en_ = enorm[ncol_];                                      \
    _Pragma("unroll")                                                    \
    for (int r = 0; r < 8; ++r) {                                        \
        float d_ = xnr[r] + en_ - 2.0f * acc_[r];                        \
        if (d_ < minv[r]) { minv[r] = d_; mini[r] = ncol_; }             \
    }                                                                    \
} while (0)

__global__ __launch_bounds__(256) void vq_main_kernel(
    const float* __restrict__ x,        // [32, 64, 64, 64] NCHW
    const float* __restrict__ cb,       // [512, 64]
    float* __restrict__ out,            // concatenated outputs
    float* __restrict__ partial)        // [NBLOCKS] loss partials
{
    extern __shared__ float smem[];
    float* cb_lds  = smem;                         // [512][65]
    float* enorm   = cb_lds + CB_FLOATS;           // [512]
    float* x_lds   = enorm + K_CODES;              // [8][16][65]
    int*   idx_lds = (int*)(x_lds + XT_FLOATS);    // [8][16]
    float* wloss   = (float*)(idx_lds + WAVES * ROWS_PER_WAVE); // [8]

    const int tid  = threadIdx.x;
    const int wave = tid >> 5;
    const int lane = tid & 31;
    const int half = lane >> 4;   // 0: lanes 0-15, 1: lanes 16-31
    const int l16  = lane & 15;

    // ---- stage codebook into LDS (padded) ----
    for (int i = tid; i < K_CODES * DIM; i += 256) {
        int k = i >> 6, c = i & 63;
        cb_lds[k * CB_STRIDE + c] = cb[i];
    }
    __syncthreads();

    // ---- ||e||^2 per code ----
    for (int k = tid; k < K_CODES; k += 256) {
        float s = 0.f;
        #pragma unroll 8
        for (int c = 0; c < DIM; ++c) { float v = cb_lds[k * CB_STRIDE + c]; s += v * v; }
        enorm[k] = s;
    }

    // ---- stage this wave's 16-row x tile (NCHW gather -> [row][chan]) ----
    const int rowBase = blockIdx.x * ROWS_PER_BLOCK + wave * ROWS_PER_WAVE;
    const int bimg = rowBase >> 12;      // 4096 rows per batch image
    const int hw0  = rowBase & (HW - 1);
    const float* xb = x + (size_t)bimg * DIM * HW;
    float* xw = x_lds + wave * ROWS_PER_WAVE * X_STRIDE;
    #pragma unroll 4
    for (int i = 0; i < 32; ++i) {
        int idx = lane + 32 * i;              // 0..1023
        int r = idx & 15, c = idx >> 4;       // 16-lane groups contiguous in n
        xw[r * X_STRIDE + c] = xb[(size_t)c * HW + hw0 + r];
    }
    if (lane < 16) {                          // row norms into col 64
        float s = 0.f;
        #pragma unroll 8
        for (int c = 0; c < DIM; ++c) { float v = xw[lane * X_STRIDE + c]; s += v * v; }
        xw[lane * X_STRIDE + DIM] = s;
    }
    __syncthreads();

    // ---- preload A fragments: 16x4 fp32 layout ----
    // lane<16: row M=l16 holds K=4kk+0,4kk+1 ; lane>=16: same row, K=4kk+2,4kk+3
    v2f afrag[16];
    #pragma unroll
    for (int kk = 0; kk < 16; ++kk) {
        int k0 = 4 * kk + 2 * half;
        afrag[kk].x = xw[l16 * X_STRIDE + k0];
        afrag[kk].y = xw[l16 * X_STRIDE + k0 + 1];
    }
    // x-norms for the 8 output rows this lane owns (C/D layout: half selects M 0-7 / 8-15)
    float xnr[8];
    #pragma unroll
    for (int r = 0; r < 8; ++r) xnr[r] = xw[(half * 8 + r) * X_STRIDE + DIM];

    float minv[8]; int mini[8];
    #pragma unroll
    for (int r = 0; r < 8; ++r) { minv[r] = 3.4e38f; mini[r] = 0; }

    // ---- GEMM + running argmin: 32 column tiles, double-buffered B fragments ----
    v2f bA[16], bB[16];
    LOAD_B(0, bA);
    for (int ct = 0; ct < 32; ct += 2) {
        LOAD_B(ct + 1, bB);       // prefetch odd tile while even tile computes
        DO_TILE(ct, bA);
        if (ct + 2 < 32) LOAD_B(ct + 2, bA);
        DO_TILE(ct + 1, bB);
    }

    // ---- reduce (min,idx) across the 16 lanes of each column group ----
    #pragma unroll
    for (int off = 1; off < 16; off <<= 1) {
        #pragma unroll
        for (int r = 0; r < 8; ++r) {
            float ov = __shfl_xor(minv[r], off, 32);
            int   oi = __shfl_xor(mini[r], off, 32);
            if (ov < minv[r] || (ov == minv[r] && oi < mini[r])) { minv[r] = ov; mini[r] = oi; }
        }
    }
    if (l16 == 0) {
        #pragma unroll
        for (int r = 0; r < 8; ++r) idx_lds[wave * 16 + half * 8 + r] = mini[r];
    }
    __syncthreads();

    // ---- indices output (as float) ----
    if (lane < 16)
        out[IDX_OFF + rowBase + lane] = (float)idx_lds[wave * 16 + lane];

    // ---- quantized output in NCHW, coalesced: 16-lane groups write 16 contiguous rows ----
    // lane -> (r = lane&15, c = c_base + lane>>4); addresses c*HW + r contiguous per group
    {
        float* outQ = out + Q_OFF + (size_t)bimg * DIM * HW + hw0;
        const int qr  = l16;
        const int idq = idx_lds[wave * 16 + qr];
        const float* cq = cb_lds + idq * CB_STRIDE;
        #pragma unroll 8
        for (int cbase = 0; cbase < DIM; cbase += 2) {
            int c = cbase + half;
            outQ[(size_t)c * HW + qr] = cq[c];
        }
    }

    // ---- one-hot encodings, float4 coalesced ----
    float* outE = out + ENC_OFF;
    for (int r = 0; r < ROWS_PER_WAVE; ++r) {
        int id = idx_lds[wave * 16 + r];
        float* er = outE + (size_t)(rowBase + r) * K_CODES;
        #pragma unroll
        for (int j = 0; j < 4; ++j) {
            int c0 = (j * 32 + lane) * 4;
            float4 v;
            v.x = (c0     == id) ? 1.f : 0.f;
            v.y = (c0 + 1 == id) ? 1.f : 0.f;
            v.z = (c0 + 2 == id) ? 1.f : 0.f;
            v.w = (c0 + 3 == id) ? 1.f : 0.f;
            *(float4*)(er + c0) = v;
        }
    }

    // ---- loss partial: sum (e[idx] - x)^2 over this wave's tile ----
    float lacc = 0.f;
    for (int r = 0; r < ROWS_PER_WAVE; ++r) {
        const float* cr = cb_lds + idx_lds[wave * 16 + r] * CB_STRIDE;
        float d0 = cr[lane]      - xw[r * X_STRIDE + lane];
        float d1 = cr[lane + 32] - xw[r * X_STRIDE + lane + 32];
        lacc += d0 * d0 + d1 * d1;
    }
    #pragma unroll
    for (int off = 1; off < 32; off <<= 1) lacc += __shfl_xor(lacc, off, 32);
    if (lane == 0) wloss[wave] = lacc;
    __syncthreads();
    if (tid == 0) {
        float s = 0.f;
        #pragma unroll
        for (int w = 0; w < WAVES; ++w) s += wloss[w];
        partial[blockIdx.x] = s;
    }
}

// deterministic fixed-order final reduction: loss = 1.25 * mean((q - x)^2)
__global__ __launch_bounds__(256) void vq_loss_reduce_kernel(
    const float* __restrict__ partial, float* __restrict__ out)
{
    __shared__ float sm[256];
    float s = 0.f;
    for (int i = threadIdx.x; i < NBLOCKS; i += 256) s += partial[i];
    sm[threadIdx.x] = s;
    __syncthreads();
    for (int off = 128; off > 0; off >>= 1) {
        if (threadIdx.x < off) sm[threadIdx.x] += sm[threadIdx.x + off];
        __syncthreads();
    }
    if (threadIdx.x == 0)
        out[0] = 1.25f * sm[0] / (float)((size_t)N_TOTAL * DIM);
}

extern "C" void kernel_launch(void* const* d_in, const int* in_sizes, int n_in,
                              void* d_out, int out_size, void* d_ws, size_t ws_size,
                              hipStream_t stream) {
    const float* x  = (const float*)d_in[0];   // [32,64,64,64] fp32
    const float* cb = (const float*)d_in[1];   // [512,64] fp32
    float* out      = (float*)d_out;
    float* partial  = (float*)d_ws;            // NBLOCKS floats

    vq_main_kernel<<<NBLOCKS, 256, SMEM_BYTES, stream>>>(x, cb, out, partial);
    vq_loss_reduce_kernel<<<1, 256, 0, stream>>>(partial, out);
}